// SimpleGCNModel_69286412419643
// MI455X (gfx1250) — compile-verified
//
#include <hip/hip_runtime.h>
#include <hip/hip_bf16.h>

#define NUM_NODES  1048576
#define NUM_FEAT   128
#define HIDDEN     64
#define NUM_GRAPHS 4096
#define OUT_DIM    5

typedef __attribute__((ext_vector_type(2))) float v2f;
typedef __attribute__((ext_vector_type(8))) float v8f;

// ---------------------------------------------------------------------------
// Kernel 0: zero the workspace accumulators (sums + counts)
// ---------------------------------------------------------------------------
__global__ void gcn_zero_kernel(float* __restrict__ p, int n) {
    int i = blockIdx.x * blockDim.x + threadIdx.x;
    if (i < n) p[i] = 0.0f;
}

// ---------------------------------------------------------------------------
// Kernel 1: segment-sum pooling over sorted batch ids.
// One wave handles 128 consecutive rows; lane owns 4 features (float4).
// Running segment accumulator in registers; global atomics only on segment
// boundary flushes (batch is sorted -> flushes are rare, branch is uniform).
// ---------------------------------------------------------------------------
#define ROWS_PER_WAVE 128
#define POOL_WAVES    8

__device__ __forceinline__ void pool_flush(float* __restrict__ sums,
                                           float* __restrict__ counts,
                                           int seg, float4 acc, float cnt, int lane) {
    float* s = sums + (size_t)seg * NUM_FEAT + lane * 4;
    atomicAdd(s + 0, acc.x);
    atomicAdd(s + 1, acc.y);
    atomicAdd(s + 2, acc.z);
    atomicAdd(s + 3, acc.w);
    if (lane == 0) atomicAdd(counts + seg, cnt);
}

__global__ __launch_bounds__(POOL_WAVES * 32)
void gcn_pool_kernel(const float* __restrict__ x,
                     const int* __restrict__ batch,
                     float* __restrict__ sums,
                     float* __restrict__ counts) {
    const int lane = threadIdx.x & 31;
    const int wave = threadIdx.x >> 5;
    const int base = (blockIdx.x * POOL_WAVES + wave) * ROWS_PER_WAVE;
    const float4* __restrict__ x4 = (const float4*)x;

    float4 acc = make_float4(0.f, 0.f, 0.f, 0.f);
    float  cnt = 0.f;
    int    cur = batch[base];

    for (int r = 0; r < ROWS_PER_WAVE; r += 4) {
        const int row0 = base + r;
        // prefetch ahead (speculative; dropped if translation fails)
        __builtin_prefetch(&x4[(size_t)(row0 + 16) * 32 + lane], 0, 3);

        float4 v0 = x4[(size_t)(row0 + 0) * 32 + lane];
        float4 v1 = x4[(size_t)(row0 + 1) * 32 + lane];
        float4 v2 = x4[(size_t)(row0 + 2) * 32 + lane];
        float4 v3 = x4[(size_t)(row0 + 3) * 32 + lane];
        int g0 = batch[row0 + 0];
        int g1 = batch[row0 + 1];
        int g2 = batch[row0 + 2];
        int g3 = batch[row0 + 3];

#define POOL_STEP(g, v)                                                   \
        if (g != cur) {                                                   \
            pool_flush(sums, counts, cur, acc, cnt, lane);                \
            cur = g; acc = v; cnt = 1.f;                                  \
        } else {                                                          \
            acc.x += v.x; acc.y += v.y; acc.z += v.z; acc.w += v.w;       \
            cnt += 1.f;                                                   \
        }
        POOL_STEP(g0, v0)
        POOL_STEP(g1, v1)
        POOL_STEP(g2, v2)
        POOL_STEP(g3, v3)
#undef POOL_STEP
    }
    pool_flush(sums, counts, cur, acc, cnt, lane);
}

// ---------------------------------------------------------------------------
// Kernel 2: fused mean-finalize + 3-layer MLP using V_WMMA_F32_16X16X4_F32.
// One wave computes a 16-graph tile end-to-end; activations bounce through a
// private 16x64 LDS slab to convert D-layout -> A-layout between layers.
//
// f32 WMMA fragment layouts (ISA 7.12.2):
//   A 16x4: lane L (<16): {K=k0, k0+1} of row M=L;  lane L(>=16): {K=k0+2, k0+3} of row M=L-16
//   B 4x16: lane L (<16): {K=k0, k0+1} of col N=L;  lane L(>=16): {K=k0+2, k0+3} of col N=L-16
//   C/D 16x16: VGPR v: lanes 0-15 -> (M=v, N=lane); lanes 16-31 -> (M=v+8, N=lane-16)
// ---------------------------------------------------------------------------
__global__ __launch_bounds__(256)
void gcn_mlp_kernel(const float* __restrict__ sums,
                    const float* __restrict__ counts,
                    const float* __restrict__ W1, const float* __restrict__ b1,
                    const float* __restrict__ W2, const float* __restrict__ b2,
                    const float* __restrict__ W3, const float* __restrict__ b3,
                    float* __restrict__ out) {
    __shared__ float lds[8][16 * HIDDEN];

    const int lane  = threadIdx.x & 31;
    const int wave  = threadIdx.x >> 5;
    const int mtile = blockIdx.x * 8 + wave;       // 0..255
    const int col   = lane & 15;                   // N (or A-row) index within tile
    const int khalf = (lane >> 4) << 1;            // 0 for lanes 0-15, 2 for lanes 16-31
    const int mrow  = ((lane >> 4) << 3);          // D-layout M base: 0 or 8
    const int m0    = mtile * 16;

    // Per-lane mean scale for the A rows this lane supplies (row = col).
    const float c     = counts[m0 + col];
    const float scale = (c > 0.f) ? (1.f / c) : 0.f;
    const float* __restrict__ arow = sums + (size_t)(m0 + col) * NUM_FEAT;
    float* __restrict__ h = lds[wave];

    // ---- Layer 1: [16x128] @ [128x64] ----
    v8f acc1[4] = {};
    for (int k0 = 0; k0 < NUM_FEAT; k0 += 4) {
        v2f a;
        a.x = arow[k0 + khalf + 0] * scale;
        a.y = arow[k0 + khalf + 1] * scale;
#pragma unroll
        for (int n = 0; n < 4; ++n) {
            v2f b;
            b.x = W1[(k0 + khalf + 0) * HIDDEN + n * 16 + col];
            b.y = W1[(k0 + khalf + 1) * HIDDEN + n * 16 + col];
            acc1[n] = __builtin_amdgcn_wmma_f32_16x16x4_f32(
                false, a, false, b, (short)0, acc1[n], false, false);
        }
    }
#pragma unroll
    for (int n = 0; n < 4; ++n) {
        const float bb = b1[n * 16 + col];
#pragma unroll
        for (int v = 0; v < 8; ++v) {
            float val = acc1[n][v] + bb;
            h[(mrow + v) * HIDDEN + n * 16 + col] = (val > 0.f) ? val : 0.f;
        }
    }
    asm volatile("s_wait_dscnt 0" ::: "memory");   // wave's LDS writes visible to all lanes

    // ---- Layer 2: [16x64] @ [64x64] ----
    v8f acc2[4] = {};
    for (int k0 = 0; k0 < HIDDEN; k0 += 4) {
        v2f a;
        a.x = h[col * HIDDEN + k0 + khalf + 0];
        a.y = h[col * HIDDEN + k0 + khalf + 1];
#pragma unroll
        for (int n = 0; n < 4; ++n) {
            v2f b;
            b.x = W2[(k0 + khalf + 0) * HIDDEN + n * 16 + col];
            b.y = W2[(k0 + khalf + 1) * HIDDEN + n * 16 + col];
            acc2[n] = __builtin_amdgcn_wmma_f32_16x16x4_f32(
                false, a, false, b, (short)0, acc2[n], false, false);
        }
    }
    asm volatile("s_wait_dscnt 0" ::: "memory");   // all layer-2 A reads done; safe to overwrite
#pragma unroll
    for (int n = 0; n < 4; ++n) {
        const float bb = b2[n * 16 + col];
#pragma unroll
        for (int v = 0; v < 8; ++v) {
            float val = acc2[n][v] + bb;
            h[(mrow + v) * HIDDEN + n * 16 + col] = (val > 0.f) ? val : 0.f;
        }
    }
    asm volatile("s_wait_dscnt 0" ::: "memory");

    // ---- Layer 3: [16x64] @ [64x5] (N padded to 16, cols >= 5 fed zeros) ----
    v8f acc3 = {};
    for (int k0 = 0; k0 < HIDDEN; k0 += 4) {
        v2f a;
        a.x = h[col * HIDDEN + k0 + khalf + 0];
        a.y = h[col * HIDDEN + k0 + khalf + 1];
        v2f b;
        b.x = (col < OUT_DIM) ? W3[(k0 + khalf + 0) * OUT_DIM + col] : 0.f;
        b.y = (col < OUT_DIM) ? W3[(k0 + khalf + 1) * OUT_DIM + col] : 0.f;
        acc3 = __builtin_amdgcn_wmma_f32_16x16x4_f32(
            false, a, false, b, (short)0, acc3, false, false);
    }
    if (col < OUT_DIM) {
        const float bb = b3[col];
#pragma unroll
        for (int v = 0; v < 8; ++v) {
            out[(size_t)(m0 + mrow + v) * OUT_DIM + col] = acc3[v] + bb;
        }
    }
}

// ---------------------------------------------------------------------------
extern "C" void kernel_launch(void* const* d_in, const int* in_sizes, int n_in,
                              void* d_out, int out_size, void* d_ws, size_t ws_size,
                              hipStream_t stream) {
    const float* x     = (const float*)d_in[0];
    const int*   batch = (const int*)d_in[1];
    const float* W1    = (const float*)d_in[2];
    const float* b1    = (const float*)d_in[3];
    const float* W2    = (const float*)d_in[4];
    const float* b2    = (const float*)d_in[5];
    const float* W3    = (const float*)d_in[6];
    const float* b3    = (const float*)d_in[7];
    float*       out   = (float*)d_out;

    float* sums   = (float*)d_ws;                        // [NUM_GRAPHS, NUM_FEAT]
    float* counts = sums + (size_t)NUM_GRAPHS * NUM_FEAT; // [NUM_GRAPHS]

    const int nz = NUM_GRAPHS * NUM_FEAT + NUM_GRAPHS;
    gcn_zero_kernel<<<(nz + 255) / 256, 256, 0, stream>>>(sums, nz);

    const int pool_blocks = NUM_NODES / (POOL_WAVES * ROWS_PER_WAVE); // 1024
    gcn_pool_kernel<<<pool_blocks, POOL_WAVES * 32, 0, stream>>>(x, batch, sums, counts);

    gcn_mlp_kernel<<<(NUM_GRAPHS / 16) / 8, 256, 0, stream>>>(
        sums, counts, W1, b1, W2, b2, W3, b3, out);
}